// SEWDAttention_43267500539969
// MI455X (gfx1250) — compile-verified
//
#include <hip/hip_runtime.h>
#include <hip/hip_bf16.h>

typedef float v2f __attribute__((ext_vector_type(2)));
typedef float v8f __attribute__((ext_vector_type(8)));
typedef unsigned int u32x4 __attribute__((ext_vector_type(4)));
typedef int i32x4 __attribute__((ext_vector_type(4)));
typedef int i32x8 __attribute__((ext_vector_type(8)));

#define SEQ   2048
#define HDIM  1024
#define NH    16
#define DH    64
#define SPAN2 1024
#define RSCALE 0.07216878364870322f   // 1/sqrt(3*64)
#define NEG_BIG (-1.0e30f)
#define KC    32                      // K-chunk staged in LDS for gemm_bias

// TDM: ROCm 7.2 exposes the 5-arg builtin; the clang-23 toolchain ships the
// TDM header and wants a 6-arg form instead -> fall back to manual loads there.
#if !__has_include(<hip/amd_detail/amd_gfx1250_TDM.h>) && \
    __has_builtin(__builtin_amdgcn_tensor_load_to_lds) && \
    __has_builtin(__builtin_amdgcn_s_wait_tensorcnt)
#define USE_TDM 1
#else
#define USE_TDM 0
#endif

static __device__ __forceinline__ v8f wmma4(v2f a, v2f b, v8f c) {
  // V_WMMA_F32_16X16X4_F32 : D = A(16x4) * B(4x16) + C(16x16), all fp32
  return __builtin_amdgcn_wmma_f32_16x16x4_f32(false, a, false, b, (short)0, c,
                                               false, false);
}

#if USE_TDM
// DMA a [rows x 64] fp32 tile (row stride 1024 dwords in memory) into LDS,
// padding +4 dwords after every 64 dwords -> LDS row stride 68 floats.
static __device__ __forceinline__ void tdm_load_tile(const float* gsrc,
                                                     void* ldst, int rows) {
  const unsigned long long ga = (unsigned long long)(uintptr_t)gsrc;
  u32x4 g0;
  g0[0] = 1u;                                   // count=1, user descriptor
  g0[1] = (unsigned)(uintptr_t)ldst;            // LDS byte address (low 32b)
  g0[2] = (unsigned)ga;                         // global_addr[31:0]
  g0[3] = (unsigned)((ga >> 32) & 0x01FFFFFFu) | (2u << 30);  // addr[56:32]|type=2
  i32x8 g1;
  g1[0] = 0x07520000;   // data_size=4B, pad_enable, interval=64dw, amount=4dw
  g1[1] = 0x04000000;   // tensor_dim0 = 1024  (bits 79:48)
  g1[2] = 0x08000000;   // tensor_dim1 = 2048  (bits 111:80)
  g1[3] = 0x00400000;   // tile_dim0   = 64    (bits 127:112)
  g1[4] = rows;         // tile_dim1   = rows  (bits 143:128)
  g1[5] = 1024;         // tensor_dim0_stride = 1024 (bits 207:160)
  g1[6] = 0; g1[7] = 0;
  i32x4 gz = {0, 0, 0, 0};                      // groups 2/3 unused (2D)
  __builtin_amdgcn_tensor_load_to_lds(g0, g1, gz, gz, 0);
}
#endif

// ---------------------------------------------------------------------------
// C[M x 1024] = A[M x 1024] @ W[1024 x 1024] + bias.
// Block = 4 waves = 64x64 output tile. W chunk staged TRANSPOSED in LDS so
// every B fragment is one contiguous ds_load_b64 into an even VGPR pair.
// ---------------------------------------------------------------------------
__global__ void gemm_bias_kernel(const float* __restrict__ A,
                                 const float* __restrict__ W,
                                 const float* __restrict__ bias,
                                 float* __restrict__ C, int M) {
  __shared__ __align__(16) float wlds[64][36];   // [col][k], pad 36
  const int lane = threadIdx.x & 31;
  const int wave = threadIdx.x >> 5;
  const int half = lane >> 4;
  const int l15  = lane & 15;
  const int nbn  = HDIM / 64;
  const int bm = blockIdx.x / nbn, bn = blockIdx.x % nbn;
  const int row0 = bm * 64 + wave * 16;
  const int col0 = bn * 64;
  const float* arow = A + (row0 + l15) * HDIM;

  v8f acc[4] = {};
  for (int k0 = 0; k0 < HDIM; k0 += KC) {
    // cooperative transposed stage of W[k0..k0+KC-1][col0..col0+63]
    for (int i = threadIdx.x; i < KC * 64; i += 128) {
      const int r = i >> 6, c = i & 63;
      wlds[c][r] = W[(k0 + r) * HDIM + col0 + c];
    }
    __syncthreads();
    if (k0 + KC < HDIM)   // prefetch next W chunk (global_prefetch_b8)
      __builtin_prefetch(&W[(k0 + KC + (threadIdx.x >> 2)) * HDIM + col0 +
                            (threadIdx.x & 3) * 16], 0, 3);
#pragma unroll
    for (int ks = 0; ks < KC; ks += 4) {
      const int kl = ks + 2 * half;
      const int kk = k0 + kl;
      v2f a; a.x = arow[kk]; a.y = arow[kk + 1];
#pragma unroll
      for (int t = 0; t < 4; ++t) {
        v2f b; b.x = wlds[t * 16 + l15][kl]; b.y = wlds[t * 16 + l15][kl + 1];
        acc[t] = wmma4(a, b, acc[t]);
      }
    }
    __syncthreads();
  }

  float bv[4];
#pragma unroll
  for (int t = 0; t < 4; ++t) bv[t] = bias[col0 + t * 16 + l15];
#pragma unroll
  for (int i = 0; i < 8; ++i) {
    const int row = row0 + i + 8 * half;
#pragma unroll
    for (int t = 0; t < 4; ++t)
      C[row * HDIM + col0 + t * 16 + l15] = acc[t][i] + bv[t];
  }
}

// ---------------------------------------------------------------------------
// C[M x N] = A[M x 64] @ B[N x 64]^T  (C2P = Q_h PK_h^T, P2C = K_h PQ_h^T).
// Block = 4 waves = 64x64 output tile; 64x64 B tile DMA'd to LDS via TDM.
// ---------------------------------------------------------------------------
__global__ void gemm_abT_kernel(const float* __restrict__ A, int lda,
                                const float* __restrict__ B, int ldb,
                                float* __restrict__ C, int ldc, int N) {
  __shared__ __align__(16) float blds[64][68];   // pad 68: conflict-free
  const int lane = threadIdx.x & 31;
  const int wave = threadIdx.x >> 5;
  const int half = lane >> 4;
  const int l15  = lane & 15;
  const int nbn  = N / 64;
  const int bm = blockIdx.x / nbn, bn = blockIdx.x % nbn;

#if USE_TDM
  if (wave == 0) {
    tdm_load_tile(B + (size_t)(bn * 64) * ldb, &blds[0][0], 64);
    __builtin_amdgcn_s_wait_tensorcnt(0);
  }
#else
  for (int i = threadIdx.x; i < 64 * 64; i += 128) {
    const int r = i >> 6, c = i & 63;
    blds[r][c] = B[(bn * 64 + r) * ldb + c];
  }
#endif
  __syncthreads();

  const float* arow = A + (bm * 64 + wave * 16 + l15) * lda;
  v8f acc[4] = {};
#pragma unroll
  for (int ks = 0; ks < 16; ++ks) {
    const int kk = 4 * ks + 2 * half;
    v2f a; a.x = arow[kk]; a.y = arow[kk + 1];
#pragma unroll
    for (int t = 0; t < 4; ++t) {
      v2f b; b.x = blds[t * 16 + l15][kk]; b.y = blds[t * 16 + l15][kk + 1];
      acc[t] = wmma4(a, b, acc[t]);
    }
  }
#pragma unroll
  for (int i = 0; i < 8; ++i) {
    const int row = bm * 64 + wave * 16 + i + 8 * half;
#pragma unroll
    for (int t = 0; t < 4; ++t)
      C[row * ldc + bn * 64 + t * 16 + l15] = acc[t][i];
  }
}

// ---------------------------------------------------------------------------
// Fused flash-style disentangled attention for one head.
// Block = 4 waves, each owning a 16-row query tile. K key block arrives via
// the Tensor Data Mover (async DMA -> LDS, padded rows); V is staged
// transposed manually; online softmax in registers.
// score(q,k) = ( QK^T + C2P[q][idx] + P2C[k][idx] ) * RSCALE,
//              idx = clip(q-k+512, 0, 1023)
// ---------------------------------------------------------------------------
__global__ void attention_kernel(const float* __restrict__ Q,
                                 const float* __restrict__ K,
                                 const float* __restrict__ V,
                                 const float* __restrict__ C2P,
                                 const float* __restrict__ P2C,
                                 const unsigned char* __restrict__ mask,
                                 float* __restrict__ CTX, int h) {
  __shared__ __align__(16) float klds[16][68];   // K block, [key n][dim kk]
  __shared__ __align__(16) float vldsT[64][20];  // V block T, [dim col][key kk]
  __shared__ float pbuf[4][16][17];              // per-wave P tile relayout
  const int lane = threadIdx.x & 31;
  const int wave = threadIdx.x >> 5;
  const int half = lane >> 4;
  const int l15  = lane & 15;
  const int qbase = (blockIdx.x * 4 + wave) * 16;
  const int hoff  = h * DH;

  // Preload Q as 16 A-fragments covering K-dim 64
  v2f qa[16];
  {
    const float* qrow = Q + (qbase + l15) * HDIM + hoff;
#pragma unroll
    for (int ks = 0; ks < 16; ++ks) {
      const int kk = 4 * ks + 2 * half;
      qa[ks].x = qrow[kk]; qa[ks].y = qrow[kk + 1];
    }
  }

  v8f acc[4] = {};            // 16 x 64 context accumulator
  float rm[8], rl[8];
#pragma unroll
  for (int i = 0; i < 8; ++i) { rm[i] = NEG_BIG; rl[i] = 0.0f; }

  for (int kb = 0; kb < SEQ / 16; ++kb) {
    const int kbase = kb * 16;
    __syncthreads();            // previous block's LDS reads complete
#if USE_TDM
    if (wave == 0)              // async DMA of K tile, EXEC-independent
      tdm_load_tile(K + (size_t)kbase * HDIM + hoff, &klds[0][0], 16);
#else
    for (int i = threadIdx.x; i < 16 * 64; i += 128) {
      const int r = i >> 6, c = i & 63;
      klds[r][c] = K[(kbase + r) * HDIM + hoff + c];
    }
#endif
    for (int i = threadIdx.x; i < 16 * 64; i += 128) {
      const int r = i >> 6, c = i & 63;
      vldsT[c][r] = V[(kbase + r) * HDIM + hoff + c];
    }
    if (kbase + 16 < SEQ) {     // prefetch next K/V key block
      const int pr = threadIdx.x & 15, pc = (threadIdx.x >> 4) << 3;
      __builtin_prefetch(&K[(kbase + 16 + pr) * HDIM + hoff + pc], 0, 3);
      __builtin_prefetch(&V[(kbase + 16 + pr) * HDIM + hoff + pc], 0, 3);
    }
#if USE_TDM
    if (wave == 0) __builtin_amdgcn_s_wait_tensorcnt(0);
#endif
    __syncthreads();

    // ---- scores: Q (16x64) @ K^T (64x16) -------------------------------
    v8f s = {};
#pragma unroll
    for (int ks = 0; ks < 16; ++ks) {
      const int kk = 4 * ks + 2 * half;
      v2f b; b.x = klds[l15][kk]; b.y = klds[l15][kk + 1];
      s = wmma4(qa[ks], b, s);
    }

    const int ncol = kbase + l15;
    float sv[8];
#pragma unroll
    for (int i = 0; i < 8; ++i) {
      const int mrow = qbase + i + 8 * half;
      int idx = mrow - ncol + 512;
      idx = idx < 0 ? 0 : (idx > SPAN2 - 1 ? SPAN2 - 1 : idx);
      const float val =
          (s[i] + C2P[mrow * SPAN2 + idx] + P2C[ncol * SPAN2 + idx]) * RSCALE;
      const bool ok = mask[mrow * SEQ + ncol] != 0;
      sv[i] = ok ? val : NEG_BIG;
    }

    // ---- online softmax (row spread over 16 lanes per half) ------------
#pragma unroll
    for (int i = 0; i < 8; ++i) {
      float mx = sv[i];
#pragma unroll
      for (int off = 1; off < 16; off <<= 1)
        mx = fmaxf(mx, __shfl_xor(mx, off, 32));
      const float mnew = fmaxf(rm[i], mx);
      const float f = __expf(rm[i] - mnew);
      const float p = (sv[i] <= -1.0e29f) ? 0.0f : __expf(sv[i] - mnew);
      float ps = p;
#pragma unroll
      for (int off = 1; off < 16; off <<= 1)
        ps += __shfl_xor(ps, off, 32);
      rl[i] = rl[i] * f + ps;
      rm[i] = mnew;
      acc[0][i] *= f; acc[1][i] *= f; acc[2][i] *= f; acc[3][i] *= f;
      pbuf[wave][i + 8 * half][l15] = p;   // C-layout -> LDS
    }

    asm volatile("s_wait_dscnt 0" ::: "memory");  // same-wave DS RAW pin

    // ---- P (16x16) @ V (16x64) -----------------------------------------
#pragma unroll
    for (int ks = 0; ks < 4; ++ks) {
      const int kk = 4 * ks + 2 * half;
      v2f a; a.x = pbuf[wave][l15][kk]; a.y = pbuf[wave][l15][kk + 1];
#pragma unroll
      for (int t = 0; t < 4; ++t) {
        v2f b; b.x = vldsT[t * 16 + l15][kk]; b.y = vldsT[t * 16 + l15][kk + 1];
        acc[t] = wmma4(a, b, acc[t]);
      }
    }
  }

  // ---- finalize ---------------------------------------------------------
#pragma unroll
  for (int i = 0; i < 8; ++i) {
    const float inv = rl[i] > 0.0f ? 1.0f / rl[i] : 0.0f;
    const int row = qbase + i + 8 * half;
    float* out = CTX + row * HDIM + hoff;
#pragma unroll
    for (int t = 0; t < 4; ++t) out[t * 16 + l15] = acc[t][i] * inv;
  }
}

// ---------------------------------------------------------------------------
// residual + LayerNorm: out = LN(O + hidden) * g + b, one row per block
// ---------------------------------------------------------------------------
__global__ void resid_ln_kernel(const float* __restrict__ O,
                                const float* __restrict__ hs,
                                const float* __restrict__ g,
                                const float* __restrict__ beta,
                                float* __restrict__ out) {
  __shared__ float red[2][8];
  const int row = blockIdx.x;
  const int t = threadIdx.x;
  float x[4];
  float s = 0.0f, s2 = 0.0f;
#pragma unroll
  for (int j = 0; j < 4; ++j) {
    const int c = t * 4 + j;
    const float v = O[row * HDIM + c] + hs[row * HDIM + c];
    x[j] = v; s += v; s2 += v * v;
  }
#pragma unroll
  for (int off = 1; off < 32; off <<= 1) {
    s  += __shfl_xor(s,  off, 32);
    s2 += __shfl_xor(s2, off, 32);
  }
  const int wave = t >> 5, lane = t & 31;
  if (lane == 0) { red[0][wave] = s; red[1][wave] = s2; }
  __syncthreads();
  if (wave == 0) {
    float a = lane < 8 ? red[0][lane] : 0.0f;
    float b = lane < 8 ? red[1][lane] : 0.0f;
#pragma unroll
    for (int off = 1; off < 8; off <<= 1) {
      a += __shfl_xor(a, off, 32);
      b += __shfl_xor(b, off, 32);
    }
    if (lane == 0) { red[0][0] = a; red[1][0] = b; }
  }
  __syncthreads();
  const float mu   = red[0][0] * (1.0f / HDIM);
  const float var  = red[1][0] * (1.0f / HDIM) - mu * mu;
  const float rstd = rsqrtf(var + 1e-5f);
#pragma unroll
  for (int j = 0; j < 4; ++j) {
    const int c = t * 4 + j;
    out[row * HDIM + c] = (x[j] - mu) * rstd * g[c] + beta[c];
  }
}

// ---------------------------------------------------------------------------
extern "C" void kernel_launch(void* const* d_in, const int* in_sizes, int n_in,
                              void* d_out, int out_size, void* d_ws,
                              size_t ws_size, hipStream_t stream) {
  const float* hs  = (const float*)d_in[0];
  const unsigned char* mask = (const unsigned char*)d_in[1];
  const float* rel = (const float*)d_in[2];
  const float* Wq  = (const float*)d_in[3];  const float* bq = (const float*)d_in[4];
  const float* Wk  = (const float*)d_in[5];  const float* bk = (const float*)d_in[6];
  const float* Wv  = (const float*)d_in[7];  const float* bv = (const float*)d_in[8];
  const float* Wo  = (const float*)d_in[9];  const float* bo = (const float*)d_in[10];
  const float* lg  = (const float*)d_in[11]; const float* lb = (const float*)d_in[12];
  float* out = (float*)d_out;

  float* ws  = (float*)d_ws;          // ~64 MB of fp32 scratch
  float* Q   = ws;                    // 2048*1024
  float* Kb  = Q   + SEQ * HDIM;      // 2048*1024
  float* Vb  = Kb  + SEQ * HDIM;      // 2048*1024
  float* PK  = Vb  + SEQ * HDIM;      // 1024*1024
  float* PQ  = PK  + SPAN2 * HDIM;    // 1024*1024
  float* CTX = PQ  + SPAN2 * HDIM;    // 2048*1024
  float* Obf = CTX + SEQ * HDIM;      // 2048*1024
  float* C2P = Obf + SEQ * HDIM;      // 2048*1024 (current head)
  float* P2C = C2P + SEQ * SPAN2;     // 2048*1024 (current head)

  const dim3 blk(128);   // 4 waves per block

  // Projections (content q/k/v and positional q/k share weights+bias)
  gemm_bias_kernel<<<(SEQ / 64) * (HDIM / 64), blk, 0, stream>>>(hs, Wq, bq, Q, SEQ);
  gemm_bias_kernel<<<(SEQ / 64) * (HDIM / 64), blk, 0, stream>>>(hs, Wk, bk, Kb, SEQ);
  gemm_bias_kernel<<<(SEQ / 64) * (HDIM / 64), blk, 0, stream>>>(hs, Wv, bv, Vb, SEQ);
  gemm_bias_kernel<<<(SPAN2 / 64) * (HDIM / 64), blk, 0, stream>>>(rel, Wk, bk, PK, SPAN2);
  gemm_bias_kernel<<<(SPAN2 / 64) * (HDIM / 64), blk, 0, stream>>>(rel, Wq, bq, PQ, SPAN2);

  // Per-head: C2P = Q_h @ PK_h^T, P2C = K_h @ PQ_h^T, fused attention
  for (int h = 0; h < NH; ++h) {
    gemm_abT_kernel<<<(SEQ / 64) * (SPAN2 / 64), blk, 0, stream>>>(
        Q + h * DH, HDIM, PK + h * DH, HDIM, C2P, SPAN2, SPAN2);
    gemm_abT_kernel<<<(SEQ / 64) * (SPAN2 / 64), blk, 0, stream>>>(
        Kb + h * DH, HDIM, PQ + h * DH, HDIM, P2C, SPAN2, SPAN2);
    attention_kernel<<<SEQ / 64, blk, 0, stream>>>(Q, Kb, Vb, C2P, P2C, mask,
                                                   CTX, h);
  }

  // Output projection + residual + LayerNorm
  gemm_bias_kernel<<<(SEQ / 64) * (HDIM / 64), blk, 0, stream>>>(CTX, Wo, bo, Obf, SEQ);
  resid_ln_kernel<<<SEQ, 256, 0, stream>>>(Obf, hs, lg, lb, out);
}